// RelativePositionalEmbedding2d_1580547968119
// MI455X (gfx1250) — compile-verified
//
#include <hip/hip_runtime.h>
#include <hip/hip_bf16.h>

typedef __attribute__((ext_vector_type(2))) float v2f;
typedef __attribute__((ext_vector_type(4))) float v4f;
typedef __attribute__((ext_vector_type(8))) float v8f;

#define HWSZ 1024   // H*W
#define CDIM 64     // EMB_DIM
#define NREL 63     // 2*32-1 valid relative offsets

// One block = 8 wave32s, handles one (bh, 16-row strip).
// Waves 0-3: Sx tile (w_emb, rel_x), n-tiles 0..3. Waves 4-7: Sy tile (h_emb, rel_y).
// Each wave: 16x16 f32 accumulator via 16 steps of v_wmma_f32_16x16x4_f32 (K=64).
__global__ __launch_bounds__(256) void relpos2d_wmma_kernel(
    const float* __restrict__ query,   // [64(bh), 64(c), 1024(i)]
    const float* __restrict__ h_emb,   // [63, 64]
    const float* __restrict__ w_emb,   // [63, 64]
    float* __restrict__ out)           // [64(bh), 1024(i), 1024(j)]
{
    __shared__ float s_tab[2][16][64];  // [0]=Sx, [1]=Sy  (col 63 is zero-pad)

    const int bh    = blockIdx.x >> 6;   // 0..63
    const int strip = blockIdx.x & 63;   // 0..63
    const int m0    = strip << 4;        // base row i of this strip

    const int tid  = threadIdx.x;
    const int wave = tid >> 5;
    const int lane = tid & 31;

    const int sel = wave >> 2;            // 0 -> Sx (w_emb), 1 -> Sy (h_emb)
    const int n0  = (wave & 3) << 4;      // n-tile base within [0,64)

    const float* __restrict__ emb = sel ? h_emb : w_emb;
    const float* __restrict__ qb  = query + (size_t)bh * CDIM * HWSZ;

    // Fragment addressing per ISA 7.12.2 (32-bit A 16x4 / B 4x16 layouts):
    const int row   = m0 + (lane & 15);   // A: M on lanes 0-15 (repeated 16-31)
    const int col   = n0 + (lane & 15);   // B: N on lanes 0-15 (repeated 16-31)
    const int khalf = (lane >> 4) << 1;   // lanes 16-31 hold K+2,K+3

    // Branch-free padding for the one invalid column (col==63):
    // clamp the address to a valid row, then scale the value by 0.
    const float cmask = (col < NREL) ? 1.0f : 0.0f;
    const int   colc  = (col < NREL) ? col : 0;
    const float* __restrict__ ebase = emb + colc * CDIM; // row `colc` of emb (r,c)

    v8f acc = {};
    #pragma unroll
    for (int k = 0; k < CDIM; k += 4) {
        const int ka = k + khalf;
        v2f a, b;
        a.x = qb[(size_t)ka * HWSZ + row];        // A[i,c] = q[c,i]
        a.y = qb[(size_t)(ka + 1) * HWSZ + row];
        // Two consecutive f32: single b64 load (8B-aligned: colc*256 + ka*4, ka even).
        const v2f e = *(const v2f*)(ebase + ka);  // B[c,r] = emb[r,c]
        b.x = e.x * cmask;
        b.y = e.y * cmask;
        // 8 args: (neg_a, A, neg_b, B, c_mod, C, reuse_a, reuse_b)
        acc = __builtin_amdgcn_wmma_f32_16x16x4_f32(
            false, a, false, b, (short)0, acc, false, false);
    }

    // Spill 16x16 D tile to LDS: VGPR v, lane L -> M = v + 8*(L>=16), N = L&15.
    {
        const int mh = (lane >> 4) << 3;
        const int nn = n0 + (lane & 15);
        #pragma unroll
        for (int v = 0; v < 8; ++v)
            s_tab[sel][v + mh][nn] = acc[v];
    }

    __syncthreads();

    // Gather phase: out[i,j] = Sx[i, xj-xi+31] + Sy[i, yj-yi+31].
    // 16 rows x 1024 cols = 4096 float4 stores; 16 per thread; fully coalesced.
    for (int it = 0; it < 16; ++it) {
        const int idx = it * 256 + tid;        // 0..4095
        const int r   = idx >> 8;              // local row 0..15
        const int j0  = (idx & 255) << 2;      // col 0..1020, step 4 (stays in one x-row)
        const int ig  = m0 + r;
        const int xi  = ig & 31, yi = ig >> 5;
        const int xj  = j0 & 31, yj = j0 >> 5;
        const int ry  = yj - yi + 31;          // in [0,62]
        const int rx  = xj - xi + 31;          // rx..rx+3 in [0,62]
        const float sy = s_tab[1][r][ry];
        v4f o;
        o.x = s_tab[0][r][rx + 0] + sy;
        o.y = s_tab[0][r][rx + 1] + sy;
        o.z = s_tab[0][r][rx + 2] + sy;
        o.w = s_tab[0][r][rx + 3] + sy;
        // Output is write-once streaming (256 MB): non-temporal store.
        __builtin_nontemporal_store(o,
            (v4f*)(out + (((size_t)bh * HWSZ + ig) << 10) + j0));
    }
}

extern "C" void kernel_launch(void* const* d_in, const int* in_sizes, int n_in,
                              void* d_out, int out_size, void* d_ws, size_t ws_size,
                              hipStream_t stream) {
    const float* query  = (const float*)d_in[0];  // [8,8,64,1024]
    const float* h_embw = (const float*)d_in[1];  // [63,64]
    const float* w_embw = (const float*)d_in[2];  // [63,64]
    // d_in[3] = rel_y, d_in[4] = rel_x : recomputed analytically in-kernel.
    float* out = (float*)d_out;                   // [8,8,1024,1024]

    (void)in_sizes; (void)n_in; (void)out_size; (void)d_ws; (void)ws_size;

    dim3 grid(64 * 64);   // 64 (b*h) x 64 strips of 16 rows
    dim3 block(256);      // 8 wave32s
    relpos2d_wmma_kernel<<<grid, block, 0, stream>>>(query, h_embw, w_embw, out);
}